// ScaleDotProductAttention_66760971649427
// MI455X (gfx1250) — compile-verified
//
#include <hip/hip_runtime.h>
#include <math.h>

// Fused attention for B=2,H=16,S=2048,D=64 (fp32), outputs (out, p_attention).
// HBM-bound on the single 512MB write of p_attention; full 16x2048 score tile
// lives in CDNA5's 320KB WGP LDS so scores are computed, softmaxed, consumed
// by PV, and streamed to HBM exactly once.

typedef __attribute__((ext_vector_type(2))) float v2f;
typedef __attribute__((ext_vector_type(8))) float v8f;

#define S_LEN    2048
#define D_HEAD   64
#define QT       16
#define NTHREADS 256

static __device__ __forceinline__ v8f wmma_f32_16x16x4(v2f a, v2f b, v8f c) {
  // 8 args: (neg_a, A, neg_b, B, c_mod, C, reuse_a, reuse_b)
  return __builtin_amdgcn_wmma_f32_16x16x4_f32(
      false, a, false, b, (short)0, c, false, false);
}

__global__ __launch_bounds__(NTHREADS, 2)
void sdpa_f32_wmma_kernel(const float* __restrict__ q,
                          const float* __restrict__ k,
                          const float* __restrict__ v,
                          float* __restrict__ outg,
                          float* __restrict__ pg) {
  __shared__ float sS[QT * S_LEN];      // 128 KB: raw scores, then exp values
  __shared__ float sRed[NTHREADS];      // reduction scratch
  __shared__ float sRowMax[QT];
  __shared__ float sRowRcp[QT];
  __shared__ float sAcc[QT * D_HEAD];   // 4 KB split-K PV partials

  const int tid  = threadIdx.x;
  const int lane = tid & 31;
  const int wave = tid >> 5;            // 8 waves (wave32)
  const int hf   = lane >> 4;           // lane half: selects K {0,1} vs {2,3}, M base 0 vs 8
  const int l    = lane & 15;           // N / M-row index within half

  const int blk = blockIdx.x;
  const int bh  = blk >> 7;             // 0..31  (B*H)
  const int qt  = blk & 127;            // 16-row query tile within S
  const int q0  = qt * QT;

  const float* Q = q + ((size_t)bh * S_LEN + q0) * D_HEAD;
  const float* K = k + (size_t)bh * S_LEN * D_HEAD;
  const float* V = v + (size_t)bh * S_LEN * D_HEAD;
  float* Og = outg + ((size_t)bh * S_LEN + q0) * D_HEAD;
  float* Pg = pg   + ((size_t)bh * S_LEN + q0) * (size_t)S_LEN;

  // ---- preload Q A-fragments: 16 K-steps of 4 cover D=64 -------------------
  // A 16x4 layout: lanes 0-15 row M=l hold K=0,1 ; lanes 16-31 hold K=2,3.
  v2f aQ[16];
  {
    const float2* qp = (const float2*)(Q + l * D_HEAD + hf * 2);
    #pragma unroll
    for (int i = 0; i < 16; ++i) {
      float2 t = qp[i * 2];             // d = i*4 + hf*2 (+0,+1), 8B aligned
      aQ[i] = (v2f){t.x, t.y};
    }
  }

  // ---- Pass 1: scores = (Q K^T) / sqrt(64) into LDS ------------------------
  for (int t = wave; t < S_LEN / 16; t += NTHREADS / 32) {
    v8f acc = {};
    // B 4x16: lane holds K[t*16+l][d0 + hf*2 + {0,1}]
    const float2* kp = (const float2*)(K + (size_t)(t * 16 + l) * D_HEAD + hf * 2);
    #pragma unroll
    for (int i = 0; i < 16; ++i) {
      float2 tb = kp[i * 2];
      acc = wmma_f32_16x16x4(aQ[i], (v2f){tb.x, tb.y}, acc);
    }
    // C/D layout: VGPR r -> M = r + hf*8, N = l
    #pragma unroll
    for (int r = 0; r < 8; ++r)
      sS[(r + hf * 8) * S_LEN + t * 16 + l] = acc[r] * 0.125f;
  }
  __syncthreads();

  // ---- Pass 2: row max ------------------------------------------------------
  const int rrow = tid >> 4;                    // 16 threads per row
  const int c0   = (tid & 15) * (S_LEN / 16);   // 128-column chunk
  float4* srow4  = (float4*)&sS[rrow * S_LEN + c0];
  {
    float m = -3.4e38f;
    #pragma unroll 4
    for (int j = 0; j < 32; ++j) {
      float4 x = srow4[j];
      m = fmaxf(m, fmaxf(fmaxf(x.x, x.y), fmaxf(x.z, x.w)));
    }
    sRed[tid] = m;
  }
  __syncthreads();
  if (tid < QT) {
    float mm = -3.4e38f;
    #pragma unroll
    for (int j = 0; j < 16; ++j) mm = fmaxf(mm, sRed[tid * 16 + j]);
    sRowMax[tid] = mm;
  }
  __syncthreads();

  // ---- Pass 3: exp in place + row sum --------------------------------------
  {
    const float mm = sRowMax[rrow];
    float sum = 0.f;
    #pragma unroll 4
    for (int j = 0; j < 32; ++j) {
      float4 x = srow4[j];
      x.x = __expf(x.x - mm); x.y = __expf(x.y - mm);
      x.z = __expf(x.z - mm); x.w = __expf(x.w - mm);
      sum += (x.x + x.y) + (x.z + x.w);
      srow4[j] = x;
    }
    sRed[tid] = sum;
  }
  __syncthreads();
  if (tid < QT) {
    float ss = 0.f;
    #pragma unroll
    for (int j = 0; j < 16; ++j) ss += sRed[tid * 16 + j];
    sRowRcp[tid] = 1.0f / ss;
  }
  __syncthreads();

  // ---- Pass 4: out = (E V) / rowsum, split-K across wave pairs -------------
  const int dt   = wave & 3;            // d-tile: columns d0..d0+15 of out
  const int kh   = wave >> 2;           // K-half: keys [kh*1024, kh*1024+1024)
  const int d0   = dt * 16;
  const int kbeg = kh * (S_LEN / 2);
  v8f acc = {};
  const float* vbase = V + d0 + l;
  #pragma unroll 4
  for (int kb = 0; kb < S_LEN / 8; ++kb) {      // 256 steps of K=4
    const int kbase = kbeg + kb * 4;
    // A 16x4 from LDS exp tile (8B-aligned ds_load_b64)
    const float2* ap = (const float2*)&sS[l * S_LEN + kbase + hf * 2];
    float2 ta = ap[0];
    // B 4x16 from V (two b32 loads, rows kbase+hf*2, kbase+hf*2+1)
    const float* vp = vbase + (size_t)(kbase + hf * 2) * D_HEAD;
    acc = wmma_f32_16x16x4((v2f){ta.x, ta.y}, (v2f){vp[0], vp[D_HEAD]}, acc);
  }
  if (kh == 1) {
    #pragma unroll
    for (int r = 0; r < 8; ++r)
      sAcc[(r + hf * 8) * D_HEAD + d0 + l] = acc[r];
  }

  // ---- Pass 5: stream normalized P to HBM once, coalesced b128 -------------
  {
    const float4* s4 = (const float4*)sS;
    float4* p4 = (float4*)Pg;
    for (int j = tid; j < QT * S_LEN / 4; j += NTHREADS) {
      const float rcp = sRowRcp[j >> 9];        // 512 float4 per row
      float4 e = s4[j];
      e.x *= rcp; e.y *= rcp; e.z *= rcp; e.w *= rcp;
      p4[j] = e;
    }
  }
  __syncthreads();

  if (kh == 0) {
    #pragma unroll
    for (int r = 0; r < 8; ++r) {
      const int M = r + hf * 8;
      const float val = acc[r] + sAcc[M * D_HEAD + d0 + l];
      Og[M * D_HEAD + d0 + l] = val * sRowRcp[M];
    }
  }
}

extern "C" void kernel_launch(void* const* d_in, const int* in_sizes, int n_in,
                              void* d_out, int out_size, void* d_ws, size_t ws_size,
                              hipStream_t stream) {
  (void)in_sizes; (void)n_in; (void)d_ws; (void)ws_size; (void)out_size;
  const float* q = (const float*)d_in[0];
  const float* k = (const float*)d_in[1];
  const float* v = (const float*)d_in[2];
  float* outg = (float*)d_out;
  float* pg = outg + (size_t)2 * 16 * 2048 * 64;   // p_attention after out
  const int nblocks = 32 * (S_LEN / QT);           // 4096
  sdpa_f32_wmma_kernel<<<dim3(nblocks), dim3(NTHREADS), 0, stream>>>(
      q, k, v, outg, pg);
}